// SelfAttentionMessageLayer_26938034880873
// MI455X (gfx1250) — compile-verified
//
#include <hip/hip_runtime.h>

typedef __attribute__((ext_vector_type(2))) float v2f;
typedef __attribute__((ext_vector_type(8))) float v8f;

#define N_NODES   50000
#define EMBED     128
#define QKV_COLS  384
#define NUM_HEADS 8
#define HEAD_DIM  16
#define N_EDGES   800000
#define NEG_FLT_MAX (-3.402823466e38f)

// ---------------------------------------------------------------------------
// Workspace init: attn[N*128]=0, denom[N*8]=0, smax[N*8]=-FLT_MAX
// ---------------------------------------------------------------------------
__global__ __launch_bounds__(256) void init_ws_kernel(float* __restrict__ smax,
                                                      float* __restrict__ denom,
                                                      float* __restrict__ attn) {
  int t = blockIdx.x * blockDim.x + threadIdx.x;
  if (t < N_NODES * EMBED) attn[t] = 0.0f;
  if (t < N_NODES * NUM_HEADS) {
    smax[t]  = NEG_FLT_MAX;
    denom[t] = 0.0f;
  }
}

// ---------------------------------------------------------------------------
// GEMM via V_WMMA_F32_16X16X4_F32:  Y[r, c] = sum_k X[r,k] * W[c,k] + bias[c]
// X: [nrows,128], W: [ncols,128] (row-major, acts as B^T), Y: [nrows, ld].
// Columns c < scale_cols get multiplied by 0.25 (q-scaling, hd^-0.5).
// One 16x16 tile per wave; 8 waves / block. K = 128 -> 32 WMMA per tile.
//
// f32 WMMA fragment layout (ISA 7.12.2):
//   A 16x4 : lane L holds row M=L&15, VGPR pair = K { 2*(L>>4), 2*(L>>4)+1 }
//   B 4x16 : lane L holds col N=L&15, VGPR pair = K { 2*(L>>4), 2*(L>>4)+1 }
//   C 16x16: VGPR g -> row g + 8*(L>>4), col L&15
// ---------------------------------------------------------------------------
__global__ __launch_bounds__(256) void gemm_wmma_kernel(
    const float* __restrict__ X, const float* __restrict__ W,
    const float* __restrict__ bias, float* __restrict__ Y,
    int nrows, int ld, int scale_cols) {
  const int lane  = threadIdx.x & 31;
  const int wave  = threadIdx.x >> 5;
  const int tileM = blockIdx.x * 8 + wave;   // 16-row tile (wave-uniform)
  if (tileM * 16 >= nrows) return;           // wave-uniform exit: EXEC stays all-1s
  const int tileN = blockIdx.y;              // 16-col tile

  const int mrow = tileM * 16 + (lane & 15); // A fragment row
  const int col  = tileN * 16 + (lane & 15); // B fragment col == C col for this lane
  const int koff = (lane >> 4) * 2;          // K sub-offset per lane half

  const float* __restrict__ arow = X + (size_t)mrow * EMBED;
  const float* __restrict__ brow = W + (size_t)col * EMBED;

  v8f c = {};
#pragma unroll 8
  for (int k = 0; k < EMBED; k += 4) {
    v2f a = *(const v2f*)(arow + k + koff);
    v2f b = *(const v2f*)(brow + k + koff);
    c = __builtin_amdgcn_wmma_f32_16x16x4_f32(false, a, false, b,
                                              (short)0, c, false, false);
  }

  const float bv    = bias[col];
  const float scale = (col < scale_cols) ? 0.25f : 1.0f;
  const int   rbase = tileM * 16 + (lane >> 4) * 8;  // C row base for this lane
#pragma unroll
  for (int g = 0; g < 8; ++g) {
    Y[(size_t)(rbase + g) * ld + col] = (c[g] + bv) * scale;
  }
}

// ---------------------------------------------------------------------------
// Per-(edge, head) score = dot16(q[tgt,h], k[src,h]); q already pre-scaled.
// atomicMax into per-(node, head) running max.
// ---------------------------------------------------------------------------
__global__ __launch_bounds__(256) void edge_scores_kernel(
    const float* __restrict__ qkv, const long long* __restrict__ edges,
    float* __restrict__ scores, float* __restrict__ smax) {
  int t = blockIdx.x * blockDim.x + threadIdx.x;
  if (t >= N_EDGES * NUM_HEADS) return;
  const int m = t >> 3;
  const int h = t & 7;
  const long long src = edges[m];
  const long long tgt = edges[N_EDGES + m];

  const float4* __restrict__ qp =
      (const float4*)(qkv + (size_t)tgt * QKV_COLS + h * HEAD_DIM);
  const float4* __restrict__ kp =
      (const float4*)(qkv + (size_t)src * QKV_COLS + EMBED + h * HEAD_DIM);

  float s = 0.0f;
#pragma unroll
  for (int i = 0; i < 4; ++i) {
    float4 a = qp[i], b = kp[i];
    s += a.x * b.x + a.y * b.y + a.z * b.z + a.w * b.w;
  }
  scores[t] = s;
  atomicMax(&smax[(size_t)tgt * NUM_HEADS + h], s);
}

// ---------------------------------------------------------------------------
// ex = exp(score - smax[tgt,h]); overwrite scores; atomicAdd into denom.
// ---------------------------------------------------------------------------
__global__ __launch_bounds__(256) void edge_exp_kernel(
    const long long* __restrict__ edges, const float* __restrict__ smax,
    float* __restrict__ scores, float* __restrict__ denom) {
  int t = blockIdx.x * blockDim.x + threadIdx.x;
  if (t >= N_EDGES * NUM_HEADS) return;
  const int m = t >> 3;
  const int h = t & 7;
  const long long tgt = edges[N_EDGES + m];
  float ex = __expf(scores[t] - smax[(size_t)tgt * NUM_HEADS + h]);
  scores[t] = ex;
  atomicAdd(&denom[(size_t)tgt * NUM_HEADS + h], ex);
}

// ---------------------------------------------------------------------------
// msg = v[src] * (ex / denom[tgt]); scatter-add into attn[tgt].
// One thread per (edge, 4 consecutive elems) -> weight shared within a head.
// ---------------------------------------------------------------------------
__global__ __launch_bounds__(256) void edge_scatter_kernel(
    const float* __restrict__ qkv, const long long* __restrict__ edges,
    const float* __restrict__ scores, const float* __restrict__ denom,
    float* __restrict__ attn) {
  int t = blockIdx.x * blockDim.x + threadIdx.x;
  if (t >= N_EDGES * 32) return;
  const int m = t >> 5;
  const int c = t & 31;      // float4 index within the 128-wide row
  const int h = c >> 2;      // 4 float4s per 16-wide head
  const long long src = edges[m];
  const long long tgt = edges[N_EDGES + m];

  const float w = scores[(size_t)m * NUM_HEADS + h] /
                  denom[(size_t)tgt * NUM_HEADS + h];
  const float4 v =
      *(const float4*)(qkv + (size_t)src * QKV_COLS + 2 * EMBED + c * 4);
  float* __restrict__ dst = attn + (size_t)tgt * EMBED + c * 4;
  atomicAdd(dst + 0, v.x * w);
  atomicAdd(dst + 1, v.y * w);
  atomicAdd(dst + 2, v.z * w);
  atomicAdd(dst + 3, v.w * w);
}

// ---------------------------------------------------------------------------
// Launch
// ---------------------------------------------------------------------------
extern "C" void kernel_launch(void* const* d_in, const int* in_sizes, int n_in,
                              void* d_out, int out_size, void* d_ws,
                              size_t ws_size, hipStream_t stream) {
  const float*     node_states = (const float*)d_in[0];
  const long long* edges       = (const long long*)d_in[1];
  const float*     qkv_w       = (const float*)d_in[2];
  const float*     qkv_b       = (const float*)d_in[3];
  const float*     out_w       = (const float*)d_in[4];
  const float*     out_b       = (const float*)d_in[5];
  float*           out         = (float*)d_out;

  // Workspace layout (all 256B-aligned):
  //   qkv    : N*384 f32 = 76.8 MB
  //   scores : M*8   f32 = 25.6 MB
  //   smax   : N*8   f32 =  1.6 MB
  //   denom  : N*8   f32 =  1.6 MB
  //   attn   : N*128 f32 = 25.6 MB
  char*  ws     = (char*)d_ws;
  float* qkv    = (float*)ws;
  float* scores = (float*)(ws + (size_t)N_NODES * QKV_COLS * 4);
  float* smax   = scores + (size_t)N_EDGES * NUM_HEADS;
  float* denom  = smax + (size_t)N_NODES * NUM_HEADS;
  float* attn   = denom + (size_t)N_NODES * NUM_HEADS;

  // 1) init accumulators
  init_ws_kernel<<<(N_NODES * EMBED + 255) / 256, 256, 0, stream>>>(smax, denom,
                                                                    attn);

  // 2) QKV projection (q pre-scaled by hd^-0.5 = 0.25 for cols < 128)
  {
    dim3 grid((N_NODES / 16 + 7) / 8, QKV_COLS / 16);
    gemm_wmma_kernel<<<grid, 256, 0, stream>>>(node_states, qkv_w, qkv_b, qkv,
                                               N_NODES, QKV_COLS, EMBED);
  }

  // 3) per-edge per-head scores + segment max
  edge_scores_kernel<<<(N_EDGES * NUM_HEADS + 255) / 256, 256, 0, stream>>>(
      qkv, edges, scores, smax);

  // 4) exp + segment sum
  edge_exp_kernel<<<(N_EDGES * NUM_HEADS + 255) / 256, 256, 0, stream>>>(
      edges, smax, scores, denom);

  // 5) weighted-value scatter-add
  edge_scatter_kernel<<<(N_EDGES * 32 + 255) / 256, 256, 0, stream>>>(
      qkv, edges, scores, denom, attn);

  // 6) output projection
  {
    dim3 grid((N_NODES / 16 + 7) / 8, EMBED / 16);
    gemm_wmma_kernel<<<grid, 256, 0, stream>>>(attn, out_w, out_b, out, N_NODES,
                                               EMBED, 0);
  }
}